// ChunkedOperation_4080218931665
// MI455X (gfx1250) — compile-verified
//
#include <hip/hip_runtime.h>

typedef __attribute__((ext_vector_type(2))) float v2f;
typedef __attribute__((ext_vector_type(4))) float v4f;
typedef __attribute__((ext_vector_type(8))) float v8f;

#define HH 16
#define NNN 8192
#define EE 64
#define CC 512
#define NCHUNK 16
#define WW (CC + (NCHUNK - 1) * 2 * CC) /* 15872 */
#define SCALE 0.125f

// XOR-swizzled LDS layout: avoids bank conflicts on the strided float2
// read-back (rows stride 512 floats == bank-period multiple otherwise).
__device__ __forceinline__ int sw_idx(int row, int col) {
    return row * CC + (col ^ ((row & 7) << 3));
}

__global__ __launch_bounds__(64) void chunked_attn_kernel(
        const float* __restrict__ Q, const float* __restrict__ K,
        const float* __restrict__ V, float* __restrict__ outO,
        float* __restrict__ outW) {
    __shared__ __align__(16) float smem[2 * 16 * CC]; // 64 KB, wave-private halves

    const int lane   = (int)(threadIdx.x & 31u);
    const int w      = (int)(threadIdx.x >> 5u);
    const int lane16 = lane & 15;
    const int hi     = lane >> 4;

    const int b    = (int)blockIdx.x;
    const int h    = b >> 8;        // 16 chunks * 16 strips per head
    const int rem  = b & 255;
    const int n    = rem >> 4;      // chunk index 0..15
    const int tile = rem & 15;      // 32-row block within chunk

    const int qb   = tile * 32 + w * 16;  // chunk-local query row base of this wave
    const int nact = (qb >> 4) + 1;       // active 16-col key tiles (causal)
    const int span = nact << 4;
    const int ksrc = (n == 0) ? 0 : (n - 1); // key/value source chunk

    float* As = smem + w * 16 * CC;

    // ---- load Q strip pre-scaled, in WMMA f32 A-operand layout ----
    const float* qrow = Q + (size_t)(h * NNN + n * CC + qb + lane16) * EE;
    v2f qa[16];
#pragma unroll
    for (int kt = 0; kt < 16; ++kt) {
        v2f t = *(const v2f*)(qrow + 4 * kt + 2 * hi);
        qa[kt] = t * SCALE;
    }

    // ---- S = scale * Q K^T over causal tiles; mask diagonal; -> LDS ----
    const float* kbase = K + (size_t)(h * NNN + ksrc * CC) * EE;
    for (int ct = 0; ct < nact; ++ct) {
        const float* krow = kbase + (size_t)(16 * ct + lane16) * EE;
        v8f acc = {0.f, 0.f, 0.f, 0.f, 0.f, 0.f, 0.f, 0.f};
#pragma unroll
        for (int kt = 0; kt < 16; ++kt) {
            v2f kb = *(const v2f*)(krow + 4 * kt + 2 * hi);
            acc = __builtin_amdgcn_wmma_f32_16x16x4_f32(
                false, qa[kt], false, kb, (short)0, acc, false, false);
        }
        const bool diag = (ct == nact - 1);
#pragma unroll
        for (int v = 0; v < 8; ++v) {
            float s = acc[v];
            const int m = v + 8 * hi;            // C-layout: row m, col lane16
            if (diag && (lane16 > m)) s = -__builtin_inff(); // keep col<=row
            As[sw_idx(m, 16 * ct + lane16)] = s;
        }
    }

    // ---- softmax over row `lane16` of the strip (cols < span) ----
    float rmax = -__builtin_inff();
    for (int c = 4 * hi; c < span; c += 8) {
        v4f s4 = *(const v4f*)&As[sw_idx(lane16, c)];
        rmax = fmaxf(rmax, fmaxf(fmaxf(s4.x, s4.y), fmaxf(s4.z, s4.w)));
    }
    rmax = fmaxf(rmax, __shfl_xor(rmax, 16));
    float rsum = 0.f;
    for (int c = 4 * hi; c < span; c += 8) {
        v4f s4 = *(const v4f*)&As[sw_idx(lane16, c)];
        v4f e4;
        e4.x = __expf(s4.x - rmax);
        e4.y = __expf(s4.y - rmax);
        e4.z = __expf(s4.z - rmax);
        e4.w = __expf(s4.w - rmax);
        *(v4f*)&As[sw_idx(lane16, c)] = e4;
        rsum += e4.x + e4.y + e4.z + e4.w;
    }
    rsum += __shfl_xor(rsum, 16);
    const float rinv = 1.0f / rsum;

    // ---- normalize; write A to LDS + global weights; zero-fill the rest ----
    const int colbase = (n == 0) ? 0 : (CC + (n - 1) * 2 * CC);
    const int zend    = (n == 0) ? CC : 2 * CC;
    float* wrow = outW + (size_t)(h * CC + qb + lane16) * WW + colbase;
    for (int c = 4 * hi; c < span; c += 8) {
        v4f e4 = *(const v4f*)&As[sw_idx(lane16, c)];
        e4 *= rinv;
        *(v4f*)&As[sw_idx(lane16, c)] = e4;
        *(v4f*)(wrow + c) = e4;
    }
    const v4f z4 = {0.f, 0.f, 0.f, 0.f};
    for (int c = span + 4 * hi; c < zend; c += 8) {
        *(v4f*)(wrow + c) = z4;
    }

    // ---- O = A @ V (A from LDS in A-operand layout, V as B-operand) ----
    const float* vbase = V + (size_t)(h * NNN + ksrc * CC) * EE;
    v8f oacc[4];
#pragma unroll
    for (int ctv = 0; ctv < 4; ++ctv)
        oacc[ctv] = (v8f){0.f, 0.f, 0.f, 0.f, 0.f, 0.f, 0.f, 0.f};
    const int nk2 = nact << 2;
    for (int kt2 = 0; kt2 < nk2; ++kt2) {
        const int k0 = 4 * kt2 + 2 * hi;
        v2f a2 = *(const v2f*)&As[sw_idx(lane16, k0)];
        const float* vr = vbase + (size_t)k0 * EE + lane16;
#pragma unroll
        for (int ctv = 0; ctv < 4; ++ctv) {
            v2f b2 = {vr[16 * ctv], vr[EE + 16 * ctv]};
            oacc[ctv] = __builtin_amdgcn_wmma_f32_16x16x4_f32(
                false, a2, false, b2, (short)0, oacc[ctv], false, false);
        }
    }

    float* orow = outO + (size_t)(h * NNN + n * CC + qb) * EE;
#pragma unroll
    for (int ctv = 0; ctv < 4; ++ctv) {
#pragma unroll
        for (int v = 0; v < 8; ++v) {
            orow[(size_t)(v + 8 * hi) * EE + 16 * ctv + lane16] = oacc[ctv][v];
        }
    }
}

extern "C" void kernel_launch(void* const* d_in, const int* in_sizes, int n_in,
                              void* d_out, int out_size, void* d_ws, size_t ws_size,
                              hipStream_t stream) {
    (void)in_sizes; (void)n_in; (void)out_size; (void)d_ws; (void)ws_size;
    const float* Q = (const float*)d_in[0];
    const float* K = (const float*)d_in[1];
    const float* V = (const float*)d_in[2];
    float* outO = (float*)d_out;
    float* outW = outO + (size_t)HH * NNN * EE;  // weights follow output, flat

    dim3 grid(HH * NCHUNK * (CC / 32)); // 4096 blocks
    dim3 block(64);                     // 2 waves (wave32)
    chunked_attn_kernel<<<grid, block, 0, stream>>>(Q, K, V, outO, outW);
}